// LinkPredict_15547781612315
// MI455X (gfx1250) — compile-verified
//
#include <hip/hip_runtime.h>

typedef __attribute__((ext_vector_type(2))) float v2f;
typedef __attribute__((ext_vector_type(8))) float v8f;

#define N_NODES 50000
#define DIM     128
#define NREL    32
#define NBASES  8
#define NEDGES  600000

// ---------------------------------------------------------------------------
// h0[n,:] = embedding[h_ids[n],:]   (float4 vectorized)
// ---------------------------------------------------------------------------
__global__ void gather_h0_kernel(const float* __restrict__ emb,
                                 const int* __restrict__ ids,
                                 float* __restrict__ h0) {
    int i = blockIdx.x * blockDim.x + threadIdx.x;          // over N*32 float4s
    if (i < N_NODES * (DIM / 4)) {
        int n = i >> 5, c = i & 31;
        ((float4*)h0)[(size_t)n * 32 + c] =
            ((const float4*)emb)[(size_t)ids[n] * 32 + c];
    }
}

// ---------------------------------------------------------------------------
// W[r,i,o] = sum_b w_comp[r,b] * bases[b,i,o]
// ---------------------------------------------------------------------------
__global__ void wcomb_kernel(const float* __restrict__ w_comp,
                             const float* __restrict__ bases,
                             float* __restrict__ W) {
    int idx = blockIdx.x * blockDim.x + threadIdx.x;        // over R*D*D
    if (idx < NREL * DIM * DIM) {
        int r   = idx >> 14;                                // / (D*D)
        int rem = idx & (DIM * DIM - 1);
        float acc = 0.f;
#pragma unroll
        for (int b = 0; b < NBASES; ++b)
            acc += w_comp[r * NBASES + b] * bases[(size_t)b * DIM * DIM + rem];
        W[idx] = acc;
    }
}

// ---------------------------------------------------------------------------
// Edge bucketing by etype
// ---------------------------------------------------------------------------
__global__ void zero32_kernel(int* counts, int* cursors) {
    int t = threadIdx.x;
    if (t < NREL) { counts[t] = 0; cursors[t] = 0; }
}

__global__ void hist_kernel(const int* __restrict__ etype, int* __restrict__ counts) {
    int i = blockIdx.x * blockDim.x + threadIdx.x;
    if (i < NEDGES) atomicAdd(&counts[etype[i]], 1);
}

__global__ void scan_kernel(const int* __restrict__ counts, int* __restrict__ poff) {
    if (threadIdx.x == 0) {
        int off = 0;
        for (int r = 0; r < NREL; ++r) {
            poff[r] = off;
            off += (counts[r] + 15) & ~15;                  // 16-align each bucket
        }
    }
}

__global__ void initperm_kernel(int* __restrict__ perm, int n) {
    int i = blockIdx.x * blockDim.x + threadIdx.x;
    if (i < n) perm[i] = -1;
}

__global__ void scatter_kernel(const int* __restrict__ etype,
                               const int* __restrict__ poff,
                               int* __restrict__ cursors,
                               int* __restrict__ perm) {
    int i = blockIdx.x * blockDim.x + threadIdx.x;
    if (i < NEDGES) {
        int r   = etype[i];
        int pos = poff[r] + atomicAdd(&cursors[r], 1);
        perm[pos] = i;
    }
}

// ---------------------------------------------------------------------------
// Self-loop GEMM: out[n,:] = h[n,:] @ loopw + bias   (WMMA f32 16x16x4)
// Persistent: B fragments (this wave's 32 columns of loopw) are hoisted into
// registers once, reused across all row tiles handled by the block.
// ---------------------------------------------------------------------------
__global__ __launch_bounds__(128)
void selfloop_kernel(const float* __restrict__ h,
                     const float* __restrict__ loopw,
                     const float* __restrict__ bias,
                     float* __restrict__ out,
                     int nRowTiles) {
    __shared__ float As[16][136];                           // pitch 136: f4-aligned
    const int tid  = threadIdx.x;
    const int wv   = tid >> 5, lane = tid & 31;
    const int hf   = lane >> 4, l = lane & 15;

    // preload B fragments + bias (loop-invariant)
    v2f bf[2][32];
    float bn[2];
#pragma unroll
    for (int ct = 0; ct < 2; ++ct) {
        const int cb = wv * 32 + ct * 16;
        bn[ct] = bias[cb + l];
#pragma unroll
        for (int ks = 0; ks < 32; ++ks) {
            const int k = ks * 4 + 2 * hf;
            bf[ct][ks].x = loopw[(size_t)k * DIM + cb + l];
            bf[ct][ks].y = loopw[(size_t)(k + 1) * DIM + cb + l];
        }
    }

    for (int t = blockIdx.x; t < nRowTiles; t += gridDim.x) {
        const int row0 = t * 16;
        {   // stage 16x128 A tile
            const int m  = tid >> 3;
            const int c0 = (tid & 7) * 16;
            const float* hrow = h + (size_t)(row0 + m) * DIM + c0;
#pragma unroll
            for (int j = 0; j < 16; j += 4)
                *(float4*)&As[m][c0 + j] = *(const float4*)(hrow + j);
        }
        __syncthreads();

#pragma unroll
        for (int ct = 0; ct < 2; ++ct) {
            const int cb = wv * 32 + ct * 16;
            v8f c = {};
#pragma unroll
            for (int ks = 0; ks < 32; ++ks) {
                v2f a = *(const v2f*)&As[l][ks * 4 + 2 * hf];
                c = __builtin_amdgcn_wmma_f32_16x16x4_f32(false, a, false, bf[ct][ks],
                                                          (short)0, c, false, false);
            }
#pragma unroll
            for (int v = 0; v < 8; ++v)
                out[(size_t)(row0 + v + 8 * hf) * DIM + cb + l] = c[v] + bn[ct];
        }
        __syncthreads();
    }
}

// ---------------------------------------------------------------------------
// Edge message GEMM with async global->LDS double-buffered A staging.
//   msg = (h[src] @ W[r]) * norm   -> atomic add into out[dst]
// grid = (NREL, tile_groups); block = 128 threads (4 wave32s).
// B fragments of W[r] live in registers; A tiles arrive via
// GLOBAL_LOAD_ASYNC_TO_LDS_B128 overlapped with the WMMA of the prior tile.
// ---------------------------------------------------------------------------
__global__ __launch_bounds__(128)
void edge_msg_kernel(const float* __restrict__ h,
                     const float* __restrict__ W,
                     const int*   __restrict__ src,
                     const int*   __restrict__ dst,
                     const float* __restrict__ norm,
                     const int*   __restrict__ perm,
                     const int*   __restrict__ counts,
                     const int*   __restrict__ poff,
                     float*       __restrict__ out) {
    __shared__ float As[2][16][136];
    __shared__ int   edst[2][16];
    __shared__ float snorm[2][16];

    const int r      = blockIdx.x;
    const int cnt    = counts[r];
    const int base   = poff[r];
    const int ntiles = (cnt + 15) >> 4;
    const int tid  = threadIdx.x;
    const int wv   = tid >> 5, lane = tid & 31;
    const int hf   = lane >> 4, l = lane & 15;
    const float* Wr = W + (size_t)r * DIM * DIM;

    // preload this wave's 32 columns of W[r] as WMMA B fragments (128 VGPRs)
    v2f bf[2][32];
#pragma unroll
    for (int ct = 0; ct < 2; ++ct) {
        const int cb = wv * 32 + ct * 16;
#pragma unroll
        for (int ks = 0; ks < 32; ++ks) {
            const int k = ks * 4 + 2 * hf;
            bf[ct][ks].x = Wr[(size_t)k * DIM + cb + l];
            bf[ct][ks].y = Wr[(size_t)(k + 1) * DIM + cb + l];
        }
    }

    // staging role of this thread: row m, 16-float column chunk c0
    const int m  = tid >> 3;
    const int c0 = (tid & 7) * 16;
    // LDS byte address of As[0][m][c0] (generic LDS addr: low 32 bits = LDS offset)
    const unsigned ldsBase = (unsigned)(size_t)&As[0][0][0];
    const unsigned myLds0  = ldsBase + (unsigned)((m * 136 + c0) * 4);
    const unsigned bufStride = (unsigned)(16 * 136 * 4);

    int t = blockIdx.y;
    int p = 0;

    if (t < ntiles) {   // prologue: kick off tile t into buffer 0
        const int e = perm[base + t * 16 + m];
        if ((tid & 7) == 0) {
            edst[0][m]  = (e >= 0) ? dst[e]  : -1;
            snorm[0][m] = (e >= 0) ? norm[e] : 0.f;
        }
        const float* hrow = h + (size_t)(e >= 0 ? src[e] : 0) * DIM + c0;
#pragma unroll
        for (int j = 0; j < 4; ++j) {
            asm volatile("global_load_async_to_lds_b128 %0, %1, off"
                         :: "v"(myLds0 + (unsigned)(j * 16)), "v"(hrow + j * 4)
                         : "memory");
        }
    }

    for (; t < ntiles; t += gridDim.y) {
        const int tn = t + gridDim.y;
        if (tn < ntiles) {  // prefetch next tile into the other buffer
            const int e = perm[base + tn * 16 + m];
            if ((tid & 7) == 0) {
                edst[1 - p][m]  = (e >= 0) ? dst[e]  : -1;
                snorm[1 - p][m] = (e >= 0) ? norm[e] : 0.f;
            }
            const float* hrow = h + (size_t)(e >= 0 ? src[e] : 0) * DIM + c0;
            const unsigned ldsN = myLds0 + (1 - p) * bufStride;
#pragma unroll
            for (int j = 0; j < 4; ++j) {
                asm volatile("global_load_async_to_lds_b128 %0, %1, off"
                             :: "v"(ldsN + (unsigned)(j * 16)), "v"(hrow + j * 4)
                             : "memory");
            }
            // current tile's 4 async loads retire first (in-order) once <=4 remain
            asm volatile("s_wait_asynccnt 0x4" ::: "memory");
        } else {
            asm volatile("s_wait_asynccnt 0x0" ::: "memory");
        }
        __syncthreads();    // all waves' async data + edst/snorm visible

#pragma unroll
        for (int ct = 0; ct < 2; ++ct) {
            const int cb = wv * 32 + ct * 16;
            v8f c = {};
#pragma unroll
            for (int ks = 0; ks < 32; ++ks) {
                v2f a = *(const v2f*)&As[p][l][ks * 4 + 2 * hf];
                c = __builtin_amdgcn_wmma_f32_16x16x4_f32(false, a, false, bf[ct][ks],
                                                          (short)0, c, false, false);
            }
#pragma unroll
            for (int v = 0; v < 8; ++v) {
                const int mm = v + 8 * hf;
                const int d  = edst[p][mm];
                if (d >= 0)
                    atomicAdd(&out[(size_t)d * DIM + cb + l], c[v] * snorm[p][mm]);
            }
        }
        __syncthreads();    // compute done before buffer p is overwritten again
        p ^= 1;
    }
}

// ---------------------------------------------------------------------------
// In-place ReLU (float4 vectorized)
// ---------------------------------------------------------------------------
__global__ void relu_kernel(float* __restrict__ x, int n4) {
    int i = blockIdx.x * blockDim.x + threadIdx.x;
    if (i < n4) {
        float4 v = ((float4*)x)[i];
        v.x = v.x > 0.f ? v.x : 0.f;
        v.y = v.y > 0.f ? v.y : 0.f;
        v.z = v.z > 0.f ? v.z : 0.f;
        v.w = v.w > 0.f ? v.w : 0.f;
        ((float4*)x)[i] = v;
    }
}

// ---------------------------------------------------------------------------
extern "C" void kernel_launch(void* const* d_in, const int* in_sizes, int n_in,
                              void* d_out, int out_size, void* d_ws, size_t ws_size,
                              hipStream_t stream) {
    (void)in_sizes; (void)n_in; (void)out_size; (void)ws_size;

    const int*   h_ids   = (const int*)  d_in[0];
    const int*   src     = (const int*)  d_in[1];
    const int*   dst     = (const int*)  d_in[2];
    const int*   etype   = (const int*)  d_in[3];
    const float* norm    = (const float*)d_in[4];
    const float* emb     = (const float*)d_in[5];
    const float* w_comp1 = (const float*)d_in[6];
    const float* bases1  = (const float*)d_in[7];
    const float* loop_w1 = (const float*)d_in[8];
    const float* bias1   = (const float*)d_in[9];
    const float* w_comp2 = (const float*)d_in[10];
    const float* bases2  = (const float*)d_in[11];
    const float* loop_w2 = (const float*)d_in[12];
    const float* bias2   = (const float*)d_in[13];
    float* out = (float*)d_out;

    // workspace layout
    float* wsW   = (float*)d_ws;                         // R*D*D      = 524288 f
    float* h0    = wsW + (size_t)NREL * DIM * DIM;       // N*D        = 6.4e6 f
    float* out1  = h0 + (size_t)N_NODES * DIM;           // N*D
    int* counts  = (int*)(out1 + (size_t)N_NODES * DIM); // 32
    int* cursors = counts + 32;                          // 32
    int* poff    = cursors + 32;                         // 64 (padded)
    int* perm    = poff + 64;                            // E + 512

    const int PERM_N = NEDGES + NREL * 16;

    // embedding gather
    gather_h0_kernel<<<(N_NODES * 32 + 255) / 256, 256, 0, stream>>>(emb, h_ids, h0);

    // bucket edges by etype (shared by both layers)
    zero32_kernel<<<1, 32, 0, stream>>>(counts, cursors);
    hist_kernel<<<(NEDGES + 255) / 256, 256, 0, stream>>>(etype, counts);
    scan_kernel<<<1, 1, 0, stream>>>(counts, poff);
    initperm_kernel<<<(PERM_N + 255) / 256, 256, 0, stream>>>(perm, PERM_N);
    scatter_kernel<<<(NEDGES + 255) / 256, 256, 0, stream>>>(etype, poff, cursors, perm);

    const int rowTiles = N_NODES / 16;                   // 3125 exact
    dim3 egrid(NREL, 48);
    const int slBlocks = 512;                            // persistent self-loop grid

    // -------- layer 1 (ReLU) --------
    wcomb_kernel<<<(NREL * DIM * DIM + 255) / 256, 256, 0, stream>>>(w_comp1, bases1, wsW);
    selfloop_kernel<<<slBlocks, 128, 0, stream>>>(h0, loop_w1, bias1, out1, rowTiles);
    edge_msg_kernel<<<egrid, 128, 0, stream>>>(h0, wsW, src, dst, norm,
                                               perm, counts, poff, out1);
    relu_kernel<<<(N_NODES * DIM / 4 + 255) / 256, 256, 0, stream>>>(out1, N_NODES * DIM / 4);

    // -------- layer 2 (no activation) --------
    wcomb_kernel<<<(NREL * DIM * DIM + 255) / 256, 256, 0, stream>>>(w_comp2, bases2, wsW);
    selfloop_kernel<<<slBlocks, 128, 0, stream>>>(out1, loop_w2, bias2, out, rowTiles);
    edge_msg_kernel<<<egrid, 128, 0, stream>>>(out1, wsW, src, dst, norm,
                                               perm, counts, poff, out);
}